// GNNEncoder_30013231464526
// MI455X (gfx1250) — compile-verified
//
#include <hip/hip_runtime.h>
#include <hip/hip_bf16.h>
#include <math.h>

// ---------------------------------------------------------------------------
// GAT (4 layers) for MI455X / gfx1250.
// GEMMs use V_WMMA_F32_16X16X4_F32 (exact fp32 matrix pipe).
// Edge aggregation is L2-resident atomic scatter (bandwidth-bound phase).
// ---------------------------------------------------------------------------

typedef float v2f __attribute__((ext_vector_type(2)));
typedef float v8f __attribute__((ext_vector_type(8)));

#define DOUT 64
#define NEG_SLOPE 0.2f

// ---------------- GEMM: H = act(X) @ W ------------------------------------
// act(v) = bias ? max(v + bias[col], 0) : v   (fuses previous layer's bias+relu)
// One wave computes a 16x16 tile of H. blockDim = 128 (4 waves = 4 col tiles).
// A-frag layout (16x4 f32): lane<16 -> M=lane, K={k,k+1}; lane>=16 -> K={k+2,k+3}
// B-frag layout (4x16 f32): lane<16 -> N=lane, K={k,k+1}; lane>=16 -> K={k+2,k+3}
// D layout: VGPR r -> row (r + 8*(lane>=16)), col lane&15.
__global__ void gat_gemm_wmma(const float* __restrict__ X,
                              const float* __restrict__ W,
                              const float* __restrict__ bias,
                              float* __restrict__ H,
                              int N, int din) {
    extern __shared__ float As[];           // [16][din+1] padded
    const int stride = din + 1;
    const int m0 = blockIdx.x * 16;

    // Cooperative, coalesced load of the 16 x din A tile with fused activation.
    for (int idx = threadIdx.x; idx < 16 * din; idx += blockDim.x) {
        int r = idx / din, c = idx % din;
        int gr = m0 + r;
        float v = 0.0f;
        if (gr < N) {
            v = X[(size_t)gr * din + c];
            if (bias) v = fmaxf(v + bias[c], 0.0f);
        }
        As[r * stride + c] = v;
    }
    __syncthreads();

    const int wave = threadIdx.x >> 5;      // 0..3 -> column tile
    const int lane = threadIdx.x & 31;
    const int lm   = lane & 15;
    const int half = lane >> 4;             // 0 or 1
    const int n0   = wave * 16;

    v8f acc = {};
    for (int k = 0; k < din; k += 4) {
        v2f a, b;
        int ka = k + 2 * half;
        a.x = As[lm * stride + ka];
        a.y = As[lm * stride + ka + 1];
        b.x = W[(size_t)ka * DOUT + n0 + lm];
        b.y = W[(size_t)(ka + 1) * DOUT + n0 + lm];
        acc = __builtin_amdgcn_wmma_f32_16x16x4_f32(
            /*neg_a=*/false, a, /*neg_b=*/false, b,
            /*c_mod=*/(short)0, acc, /*reuse_a=*/false, /*reuse_b=*/false);
    }

#pragma unroll
    for (int r = 0; r < 8; ++r) {
        int row = m0 + r + 8 * half;
        if (row < N) H[(size_t)row * DOUT + n0 + lm] = acc[r];
    }
}

// ---------------- per-node attention logits --------------------------------
__global__ void gat_alpha(const float* __restrict__ H,
                          const float* __restrict__ wsrc,
                          const float* __restrict__ wdst,
                          float* __restrict__ asrc,
                          float* __restrict__ adst, int N) {
    int n = blockIdx.x * blockDim.x + threadIdx.x;
    if (n >= N) return;
    float s1 = 0.0f, s2 = 0.0f;
    const float* h = H + (size_t)n * DOUT;
#pragma unroll
    for (int c = 0; c < DOUT; ++c) {
        float hv = h[c];
        s1 += hv * wsrc[c];
        s2 += hv * wdst[c];
    }
    asrc[n] = s1;
    adst[n] = s2;
}

// ---------------- per-layer init ------------------------------------------
__global__ void gat_init(float* __restrict__ OUT, float* __restrict__ m,
                         float* __restrict__ s, int N) {
    int i = blockIdx.x * blockDim.x + threadIdx.x;
    if (i < N * DOUT) OUT[i] = 0.0f;
    if (i < N) { m[i] = -INFINITY; s[i] = 0.0f; }
}

// float atomic max via IEEE int-ordering trick (native int atomics in L2)
__device__ __forceinline__ void atomicMaxFloat(float* addr, float val) {
    if (val >= 0.0f)
        atomicMax((int*)addr, __float_as_int(val));
    else
        atomicMin((unsigned int*)addr, __float_as_uint(val));
}

__device__ __forceinline__ void edge_pair(const int* __restrict__ ei, int i,
                                          int e_raw, int& src, int& dst) {
    if (i < e_raw) { src = ei[i]; dst = ei[e_raw + i]; }
    else           { src = dst = i - e_raw; }          // self-loops
}

// pass 1: e = leaky_relu(asrc[src]+adst[dst]); m[dst] = max(m[dst], e)
__global__ void gat_edge1(const int* __restrict__ ei,
                          const float* __restrict__ asrc,
                          const float* __restrict__ adst,
                          float* __restrict__ ebuf, float* __restrict__ m,
                          int e_raw, int e_tot) {
    int i = blockIdx.x * blockDim.x + threadIdx.x;
    if (i >= e_tot) return;
    int src, dst;
    edge_pair(ei, i, e_raw, src, dst);
    float ev = asrc[src] + adst[dst];
    ev = ev > 0.0f ? ev : NEG_SLOPE * ev;
    ebuf[i] = ev;
    atomicMaxFloat(&m[dst], ev);
}

// pass 2: p = exp(e - m[dst]); s[dst] += p
__global__ void gat_edge2(const int* __restrict__ ei,
                          const float* __restrict__ ebuf,
                          const float* __restrict__ m,
                          float* __restrict__ pbuf, float* __restrict__ s,
                          int e_raw, int e_tot) {
    int i = blockIdx.x * blockDim.x + threadIdx.x;
    if (i >= e_tot) return;
    int src, dst;
    edge_pair(ei, i, e_raw, src, dst);
    float pv = expf(ebuf[i] - m[dst]);
    pbuf[i] = pv;
    atomicAdd(&s[dst], pv);
}

// pass 3: OUT[dst][c] += (p / (s[dst]+eps)) * H[src][c]   (64 threads / edge)
__global__ void gat_edge3(const int* __restrict__ ei,
                          const float* __restrict__ pbuf,
                          const float* __restrict__ s,
                          const float* __restrict__ H,
                          float* __restrict__ OUT,
                          int e_raw, int e_tot) {
    unsigned t = blockIdx.x * blockDim.x + threadIdx.x;
    unsigned i = t >> 6;                 // edge
    unsigned c = t & 63u;                // channel (wave covers 32 contiguous)
    if (i >= (unsigned)e_tot) return;
    int src, dst;
    edge_pair(ei, (int)i, e_raw, src, dst);
    float alpha = pbuf[i] / (s[dst] + 1e-16f);
    atomicAdd(&OUT[(size_t)dst * DOUT + c], alpha * H[(size_t)src * DOUT + c]);
}

// ---------------- final mean + bias ---------------------------------------
// block c (64 blocks) reduces channel c over all nodes: d_out[c]=mean+b4[c]
__global__ void gat_final(const float* __restrict__ OUT,
                          const float* __restrict__ b4,
                          float* __restrict__ out, int N) {
    __shared__ float sd[256];
    int c = blockIdx.x;
    float acc = 0.0f;
    for (int n = threadIdx.x; n < N; n += blockDim.x)
        acc += OUT[(size_t)n * DOUT + c];
    sd[threadIdx.x] = acc;
    __syncthreads();
    for (int off = 128; off > 0; off >>= 1) {
        if (threadIdx.x < off) sd[threadIdx.x] += sd[threadIdx.x + off];
        __syncthreads();
    }
    if (threadIdx.x == 0) out[c] = sd[0] / (float)N + b4[c];
}

// ---------------------------------------------------------------------------
extern "C" void kernel_launch(void* const* d_in, const int* in_sizes, int n_in,
                              void* d_out, int out_size, void* d_ws, size_t ws_size,
                              hipStream_t stream) {
    const float* x  = (const float*)d_in[0];
    const int*   ei = (const int*)d_in[1];          // int32 (JAX default-x64-off)
    const int din1  = in_sizes[2] / DOUT;           // 128
    const int N     = in_sizes[0] / din1;           // 50000
    const int e_raw = in_sizes[1] / 2;              // 800000
    const int e_tot = e_raw + N;                    // + self-loops

    // workspace layout (floats)
    float* H    = (float*)d_ws;
    float* OUT  = H    + (size_t)N * DOUT;
    float* asrc = OUT  + (size_t)N * DOUT;
    float* adst = asrc + N;
    float* mbuf = adst + N;
    float* sbuf = mbuf + N;
    float* ebuf = sbuf + N;
    float* pbuf = ebuf + e_tot;

    const float* Wl[4], *al_src[4], *al_dst[4], *bl[4];
    for (int l = 0; l < 4; ++l) {
        Wl[l]     = (const float*)d_in[2 + 4 * l];
        al_src[l] = (const float*)d_in[3 + 4 * l];
        al_dst[l] = (const float*)d_in[4 + 4 * l];
        bl[l]     = (const float*)d_in[5 + 4 * l];
    }

    const int nodeBlocks = (N + 255) / 256;
    const int edgeBlocks = (e_tot + 255) / 256;
    const int e3Blocks   = (int)(((long long)e_tot * DOUT + 255) / 256);

    for (int l = 0; l < 4; ++l) {
        const float* Xin  = (l == 0) ? x : OUT;
        const int    din  = (l == 0) ? din1 : DOUT;
        const float* bias = (l == 0) ? nullptr : bl[l - 1]; // fused relu(prev+b)
        size_t smem = (size_t)16 * (din + 1) * sizeof(float);

        gat_gemm_wmma<<<(N + 15) / 16, 128, smem, stream>>>(Xin, Wl[l], bias, H, N, din);
        gat_alpha<<<nodeBlocks, 256, 0, stream>>>(H, al_src[l], al_dst[l], asrc, adst, N);
        gat_init<<<(N * DOUT + 255) / 256, 256, 0, stream>>>(OUT, mbuf, sbuf, N);
        gat_edge1<<<edgeBlocks, 256, 0, stream>>>(ei, asrc, adst, ebuf, mbuf, e_raw, e_tot);
        gat_edge2<<<edgeBlocks, 256, 0, stream>>>(ei, ebuf, mbuf, pbuf, sbuf, e_raw, e_tot);
        gat_edge3<<<e3Blocks, 256, 0, stream>>>(ei, pbuf, sbuf, H, OUT, e_raw, e_tot);
    }

    gat_final<<<DOUT, 256, 0, stream>>>(OUT, bl[3], (float*)d_out, N);
}